// GELU180_39857296507323
// MI455X (gfx1250) — compile-verified
//
#include <hip/hip_runtime.h>
#include <math.h>

// Problem constants from the reference: B=2, T=1024, D=256 (fp32).
#define TT 1024
#define DD 256

typedef float v2f __attribute__((ext_vector_type(2)));
typedef float v8f __attribute__((ext_vector_type(8)));

__device__ __forceinline__ float gelu_tanh(float x) {
    const float c = 0.7978845608028654f;  // sqrt(2/pi)
    float inner = c * (x + 0.044715f * x * x * x);
    return 0.5f * x * (1.0f + tanhf(inner));
}

// ---------------------------------------------------------------------------
// Kernel 0: zero the per-(b,t) surprise accumulator (B*T floats in d_ws).
// ---------------------------------------------------------------------------
__global__ void zero_acc_kernel(float* acc, int n) {
    int i = blockIdx.x * blockDim.x + threadIdx.x;
    if (i < n) acc[i] = 0.0f;
}

// ---------------------------------------------------------------------------
// Kernel 1: causal pairwise rank surprise.
// One workgroup (1024 threads) per (b, d) column. Column cached in LDS;
// thread i counts strict "earlier-and-less" refs via a float4-unrolled
// broadcast LDS loop, then atomically accumulates |2*rank-1| into acc[b*T+t].
// ---------------------------------------------------------------------------
__global__ __launch_bounds__(TT) void rank_surprise_kernel(
    const float* __restrict__ x, float* __restrict__ acc, int B) {
    __shared__ float sh[TT];

    int col = blockIdx.x;          // 0 .. B*D-1
    int b = col / DD;
    int d = col - b * DD;
    int t = threadIdx.x;           // 0 .. T-1

    float xi = x[((size_t)(b * TT + t)) * DD + d];
    sh[t] = xi;
    __syncthreads();

    int cnt = 0;
    int j = 0;
    int jm = t & ~3;
    for (; j < jm; j += 4) {
        float4 v = *reinterpret_cast<const float4*>(&sh[j]);  // uniform ds_load_b128
        cnt += (v.x < xi);
        cnt += (v.y < xi);
        cnt += (v.z < xi);
        cnt += (v.w < xi);
    }
    for (; j < t; ++j) cnt += (sh[j] < xi);

    if (t > 0) {
        float rank = (float)cnt / (float)t;       // n_refs = t for t >= 1
        float surp = fabsf(2.0f * rank - 1.0f);
        atomicAdd(&acc[b * TT + t], surp);
    }
    // t == 0: reference forces rank = 0.5 -> contribution 0 (accumulator stays 0)
}

// ---------------------------------------------------------------------------
// Kernel 2: gelu + cosine gate + surprise gate, fused output.
// One wave (32 lanes) per 16 rows (row = b*T + t). The two per-row length-256
// reductions (sum g*ema and sum g*g) are computed with V_WMMA_F32_16X16X4_F32
// using A = ones so D[m,n] = sum_k B[k,n]: exact f32, and invariant to the
// exact K-slot assignment of the B operand layout. Every lane ends with the
// row-n (= lane%16) dot in each accumulator element.
// ---------------------------------------------------------------------------
__global__ __launch_bounds__(32) void gate_kernel(
    const float* __restrict__ x, const float* __restrict__ ema,
    const float* __restrict__ p_log_tau, const float* __restrict__ p_log_sigma,
    const float* __restrict__ p_log_w, const float* __restrict__ acc,
    float* __restrict__ out) {
    __shared__ float g_lds[16 * DD];
    __shared__ float gate_lds[16];

    int lane = threadIdx.x;        // 0..31, full wave active (EXEC all ones)
    int n = lane & 15;             // row within this 16-row tile
    int c0 = (lane < 16) ? 0 : 2;  // which pair of K-offsets this lane feeds
    int r0 = blockIdx.x * 16;      // first global row of tile
    int row = r0 + n;
    const float* xrow = x + (size_t)row * DD;

    v8f cge = {};   // accumulates sum_d g*ema
    v8f cgg = {};   // accumulates sum_d g*g
    v2f aones; aones[0] = 1.0f; aones[1] = 1.0f;

    for (int s = 0; s < DD / 4; ++s) {
        int d0 = 4 * s + c0;
        float x0 = xrow[d0];
        float x1 = xrow[d0 + 1];
        float g0 = gelu_tanh(x0);
        float g1 = gelu_tanh(x1);
        g_lds[n * DD + d0]     = g0;
        g_lds[n * DD + d0 + 1] = g1;

        v2f bge, bgg;
        bge[0] = g0 * ema[d0];
        bge[1] = g1 * ema[d0 + 1];
        bgg[0] = g0 * g0;
        bgg[1] = g1 * g1;

        // D = A(ones, 16x4) x B(4x16) + C  -> column sums of B, exact f32
        cge = __builtin_amdgcn_wmma_f32_16x16x4_f32(
            false, aones, false, bge, (short)0, cge, false, false);
        cgg = __builtin_amdgcn_wmma_f32_16x16x4_f32(
            false, aones, false, bgg, (short)0, cgg, false, false);
    }

    float dot_ge = cge[0];   // dot for row n (same in every accumulator VGPR)
    float dot_gg = cgg[0];

    // ||ema||: wave-level reduction over D=256 elements
    float es = 0.0f;
    for (int dd = lane; dd < DD; dd += 32) {
        float e = ema[dd];
        es += e * e;
    }
    for (int off = 16; off > 0; off >>= 1) es += __shfl_xor(es, off, 32);
    float ema_inv = 1.0f / fmaxf(sqrtf(es), 1e-12f);

    float tau   = expf(p_log_tau[0]);
    float sigma = log1pf(expf(p_log_sigma[0]));   // softplus
    float w     = log1pf(expf(p_log_w[0]));       // softplus

    if (lane < 16) {
        float g_inv = 1.0f / fmaxf(sqrtf(dot_gg), 1e-12f);
        float cs = dot_ge * g_inv * ema_inv;
        cs = fminf(fmaxf(cs, -1.0f), 1.0f);
        float gate_cos = expf(-tau * cs);
        float mean_surp = acc[row] * (1.0f / (float)DD);
        float surprise = tanhf(sigma * mean_surp);
        gate_lds[n] = gate_cos * (1.0f + w * surprise);
    }
    __syncthreads();

    // Coalesced write: out = g * gate[row]
    for (int e = lane; e < 16 * DD; e += 32) {
        int rr = e >> 8;        // e / DD
        int dd = e & (DD - 1);  // e % DD
        out[(size_t)(r0 + rr) * DD + dd] = g_lds[e] * gate_lds[rr];
    }
}

// ---------------------------------------------------------------------------
// Launch
// ---------------------------------------------------------------------------
extern "C" void kernel_launch(void* const* d_in, const int* in_sizes, int n_in,
                              void* d_out, int out_size, void* d_ws, size_t ws_size,
                              hipStream_t stream) {
    const float* x        = (const float*)d_in[0];
    const float* ema      = (const float*)d_in[1];
    const float* log_tau  = (const float*)d_in[2];
    const float* log_sig  = (const float*)d_in[3];
    const float* log_w    = (const float*)d_in[4];
    float* out = (float*)d_out;

    int D = in_sizes[1];                 // 256
    int BT = in_sizes[0] / D;            // B*T = 2048
    int B = BT / TT;                     // 2

    float* acc = (float*)d_ws;           // B*T floats of scratch

    zero_acc_kernel<<<(BT + 255) / 256, 256, 0, stream>>>(acc, BT);
    rank_surprise_kernel<<<B * DD, TT, 0, stream>>>(x, acc, B);
    gate_kernel<<<BT / 16, 32, 0, stream>>>(x, ema, log_tau, log_sig, log_w,
                                            acc, out);
}